// GraphVAE_83236466197161
// MI455X (gfx1250) — compile-verified
//
#include <hip/hip_runtime.h>
#include <hip/hip_bf16.h>
#include <math.h>

#define N_NODES 100000
#define N_EDGES 1600000
#define IN_DIM  128
#define HID_DIM 64
#define Z_DIM   32

typedef __attribute__((ext_vector_type(2))) float v2f;
typedef __attribute__((ext_vector_type(8))) float v8f;

// ---------------------------------------------------------------------------
// Small utility kernels
// ---------------------------------------------------------------------------
__global__ void zero_f32(float* __restrict__ p, long long n) {
  long long i = (long long)blockIdx.x * blockDim.x + threadIdx.x;
  if (i < n) p[i] = 0.0f;
}

__global__ void init_acc_deg(float* __restrict__ acc, float* __restrict__ deg) {
  int i = blockIdx.x * blockDim.x + threadIdx.x;
  if (i < 2) acc[i] = 0.0f;
  if (i < N_NODES) deg[i] = 1.0f;  // self-loop contribution
}

__global__ void deg_accum(const int* __restrict__ dst, float* __restrict__ deg) {
  int e = blockIdx.x * blockDim.x + threadIdx.x;
  if (e < N_EDGES) atomicAdd(&deg[dst[e]], 1.0f);
}

__global__ void deg_to_dinv(float* __restrict__ deg) {
  int i = blockIdx.x * blockDim.x + threadIdx.x;
  if (i < N_NODES) deg[i] = rsqrtf(deg[i]);  // deg >= 1 always
}

// ---------------------------------------------------------------------------
// f32 GEMM via V_WMMA_F32_16X16X4_F32.
// C[M x COLS] = A[M x K] * B[K x COLS], all row-major, M multiple of
// (waves_per_block/colTiles)*16, K multiple of 4, COLS multiple of 16.
// One wave owns one 16x16 output tile; EXEC is all-ones (exact grid fit).
// A 16x4 fragment: lanes 0-15 -> K={k,k+1}, lanes 16-31 -> K={k+2,k+3}.
// B 4x16 fragment: same K split, N = lane&15.
// C/D: VGPR v holds M=v (lanes 0-15) and M=v+8 (lanes 16-31).
// ---------------------------------------------------------------------------
template <int COLS>
__global__ void gemm_wmma_f32(const float* __restrict__ A,
                              const float* __restrict__ B,
                              float* __restrict__ C, int K) {
  const int lane = threadIdx.x & 31;
  const int w    = threadIdx.x >> 5;
  const int colTiles = COLS / 16;
  const int rowTilesPerBlock = (blockDim.x >> 5) / colTiles;
  const int rowTile = blockIdx.x * rowTilesPerBlock + (w / colTiles);
  const int colTile = w % colTiles;

  const int mrow  = rowTile * 16 + (lane & 15);
  const int col   = colTile * 16 + (lane & 15);
  const int khalf = (lane >> 4) << 1;  // 0 for lanes 0-15, 2 for lanes 16-31

  v8f c = {};
  for (int kk = 0; kk < K; kk += 4) {
    const int ka = kk + khalf;
    v2f a, b;
    a.x = A[(long long)mrow * K + ka];
    a.y = A[(long long)mrow * K + ka + 1];
    b.x = B[(long long)ka * COLS + col];
    b.y = B[(long long)(ka + 1) * COLS + col];
    c = __builtin_amdgcn_wmma_f32_16x16x4_f32(
        /*neg_a=*/false, a, /*neg_b=*/false, b,
        /*c_mod=*/(short)0, c, /*reuse_a=*/false, /*reuse_b=*/false);
  }

  const int rbase = rowTile * 16 + ((lane >> 4) << 3);
#pragma unroll
  for (int v = 0; v < 8; ++v) {
    C[(long long)(rbase + v) * COLS + col] = c[v];
  }
}

// ---------------------------------------------------------------------------
// Normalized edge scatter: out[dst] += in[src] * dinv[src]*dinv[dst]
// One thread per (edge, feature). dim = 1<<shift (64 or 32).
// dinv (400 KB) and the node tensors are L2-resident on MI455X (192 MB L2).
// ---------------------------------------------------------------------------
__global__ void scatter_norm(const int* __restrict__ src,
                             const int* __restrict__ dst,
                             const float* __restrict__ dinv,
                             const float* __restrict__ in,
                             float* __restrict__ out, int shift) {
  long long idx = (long long)blockIdx.x * blockDim.x + threadIdx.x;
  const int dim = 1 << shift;
  long long e = idx >> shift;
  int d = (int)(idx & (dim - 1));
  if (e < N_EDGES) {
    int s = src[e], t = dst[e];
    float norm = dinv[s] * dinv[t];
    atomicAdd(&out[(long long)t * dim + d], in[(long long)s * dim + d] * norm);
  }
}

// h = relu(agg + hW*dinv^2 + b1)   (self-loop term folded in, in place)
__global__ void finish_h(float* __restrict__ agg, const float* __restrict__ hW,
                         const float* __restrict__ dinv,
                         const float* __restrict__ b1) {
  long long idx = (long long)blockIdx.x * blockDim.x + threadIdx.x;
  if (idx < (long long)N_NODES * HID_DIM) {
    int n = (int)(idx >> 6);
    int d = (int)(idx & 63);
    float di = dinv[n];
    float v = agg[idx] + hW[idx] * di * di + b1[d];
    agg[idx] = fmaxf(v, 0.0f);
  }
}

// mu/logvar finalize + reparameterize + fused KL partial reduction.
__global__ void finish_z_kl(const float* __restrict__ aggMu,
                            const float* __restrict__ aggLv,
                            const float* __restrict__ muW,
                            const float* __restrict__ lvW,
                            const float* __restrict__ dinv,
                            const float* __restrict__ bmu,
                            const float* __restrict__ blv,
                            const float* __restrict__ eps,
                            float* __restrict__ z, float* __restrict__ acc) {
  __shared__ float sdata[256];
  long long idx = (long long)blockIdx.x * blockDim.x + threadIdx.x;
  float klt = 0.0f;
  if (idx < (long long)N_NODES * Z_DIM) {
    int n = (int)(idx >> 5);
    int d = (int)(idx & 31);
    float di = dinv[n];
    float di2 = di * di;
    float mu = aggMu[idx] + muW[idx] * di2 + bmu[d];
    float lv = aggLv[idx] + lvW[idx] * di2 + blv[d];
    z[idx] = mu + eps[idx] * expf(0.5f * lv);
    klt = 1.0f + lv - mu * mu - expf(lv);
  }
  sdata[threadIdx.x] = klt;
  __syncthreads();
  for (int s = 128; s > 0; s >>= 1) {
    if ((int)threadIdx.x < s) sdata[threadIdx.x] += sdata[threadIdx.x + s];
    __syncthreads();
  }
  if (threadIdx.x == 0) atomicAdd(&acc[1], sdata[0]);
}

// Inner-product decode + BCE-with-logits partial sums.
// One wave32 per edge (lane = z-dim), 8 edges per wave, one atomic per block.
__global__ void decode_bce(const int* __restrict__ pos_src,
                           const int* __restrict__ pos_dst,
                           const int* __restrict__ neg_src,
                           const int* __restrict__ neg_dst,
                           const float* __restrict__ z,
                           float* __restrict__ acc) {
  __shared__ float sdata[8];
  const int lane = threadIdx.x & 31;
  const int w    = threadIdx.x >> 5;
  const long long e0 = ((long long)blockIdx.x * 8 + w) * 8;
  float sum = 0.0f;
#pragma unroll
  for (int i = 0; i < 8; ++i) {
    long long e = e0 + i;  // e in [0, 2*N_EDGES)
    const int* sp; const int* dp; float label; long long ee;
    if (e < N_EDGES) { sp = pos_src; dp = pos_dst; label = 1.0f; ee = e; }
    else             { sp = neg_src; dp = neg_dst; label = 0.0f; ee = e - N_EDGES; }
    int s = sp[ee], t = dp[ee];
    float p = z[(long long)s * Z_DIM + lane] * z[(long long)t * Z_DIM + lane];
    for (int m = 16; m >= 1; m >>= 1) p += __shfl_xor(p, m, 32);
    if (lane == 0) {
      float l = p;
      sum += fmaxf(l, 0.0f) - l * label + log1pf(expf(-fabsf(l)));
    }
  }
  if (lane == 0) sdata[w] = sum;
  __syncthreads();
  if (threadIdx.x == 0) {
    float t = 0.0f;
#pragma unroll
    for (int i = 0; i < 8; ++i) t += sdata[i];
    atomicAdd(&acc[0], t);
  }
}

__global__ void final_out(const float* __restrict__ acc, float* __restrict__ out) {
  if (threadIdx.x == 0 && blockIdx.x == 0) {
    float recon = acc[0] / (2.0f * (float)N_EDGES);
    float kl = -0.5f * acc[1] / ((float)N_NODES * (float)Z_DIM);
    out[0] = recon + kl;
    out[1] = recon;
    out[2] = kl;
  }
}

// ---------------------------------------------------------------------------
// Launch
// ---------------------------------------------------------------------------
extern "C" void kernel_launch(void* const* d_in, const int* in_sizes, int n_in,
                              void* d_out, int out_size, void* d_ws, size_t ws_size,
                              hipStream_t stream) {
  const float* x   = (const float*)d_in[0];
  const float* eps = (const float*)d_in[1];
  const float* W1  = (const float*)d_in[2];
  const float* b1  = (const float*)d_in[3];
  const float* Wmu = (const float*)d_in[4];
  const float* bmu = (const float*)d_in[5];
  const float* Wlv = (const float*)d_in[6];
  const float* blv = (const float*)d_in[7];
  const int* ei  = (const int*)d_in[8];   // [2, E]
  const int* nei = (const int*)d_in[9];   // [2, E]
  const int* src = ei;
  const int* dst = ei + N_EDGES;
  const int* nsrc = nei;
  const int* ndst = nei + N_EDGES;
  float* out = (float*)d_out;

  // Workspace layout (floats)
  float* ws   = (float*)d_ws;
  float* acc  = ws;                          // 2 floats (recon_sum, kl_sum)
  float* dinv = ws + 64;                     // N
  float* buf1 = dinv + 100096;               // N*64 : hW -> (muW, lvW)
  float* buf2 = buf1 + (long long)N_NODES * 64;  // N*64 : agg1/h -> (aggMu, aggLv)
  float* zbuf = buf2 + (long long)N_NODES * 64;  // N*32
  float* muW  = buf1;
  float* lvW  = buf1 + (long long)N_NODES * Z_DIM;
  float* aggMu = buf2;
  float* aggLv = buf2 + (long long)N_NODES * Z_DIM;

  const long long NH = (long long)N_NODES * HID_DIM;  // 6.4M

  // --- degrees -> dinv ---
  init_acc_deg<<<(N_NODES + 255) / 256, 256, 0, stream>>>(acc, dinv);
  deg_accum<<<N_EDGES / 256, 256, 0, stream>>>(dst, dinv);
  deg_to_dinv<<<(N_NODES + 255) / 256, 256, 0, stream>>>(dinv);

  // --- layer 1: hW = x @ W1 (WMMA f32), propagate, relu ---
  // 256 threads = 8 waves; 4 col tiles -> 2 row tiles/block; 100000/32 = 3125
  gemm_wmma_f32<HID_DIM><<<3125, 256, 0, stream>>>(x, W1, buf1, IN_DIM);
  zero_f32<<<(unsigned)(NH / 256), 256, 0, stream>>>(buf2, NH);
  scatter_norm<<<(unsigned)((long long)N_EDGES * 64 / 256), 256, 0, stream>>>(
      src, dst, dinv, buf1, buf2, 6);
  finish_h<<<(unsigned)(NH / 256), 256, 0, stream>>>(buf2, buf1, dinv, b1);

  // --- layer 2: muW/lvW = h @ {Wmu,Wlv} (WMMA f32) ---
  // 128 threads = 4 waves; 2 col tiles -> 2 row tiles/block; 3125 blocks
  gemm_wmma_f32<Z_DIM><<<3125, 128, 0, stream>>>(buf2, Wmu, muW, HID_DIM);
  gemm_wmma_f32<Z_DIM><<<3125, 128, 0, stream>>>(buf2, Wlv, lvW, HID_DIM);

  // h (buf2) is now dead -> reuse as aggregation buffers
  zero_f32<<<(unsigned)(NH / 256), 256, 0, stream>>>(buf2, NH);
  scatter_norm<<<(unsigned)((long long)N_EDGES * 32 / 256), 256, 0, stream>>>(
      src, dst, dinv, muW, aggMu, 5);
  scatter_norm<<<(unsigned)((long long)N_EDGES * 32 / 256), 256, 0, stream>>>(
      src, dst, dinv, lvW, aggLv, 5);

  // --- z = mu + eps*exp(0.5*logvar), fused KL partials ---
  finish_z_kl<<<(unsigned)((long long)N_NODES * Z_DIM / 256), 256, 0, stream>>>(
      aggMu, aggLv, muW, lvW, dinv, bmu, blv, eps, zbuf, acc);

  // --- decode: 3.2M edges, 1 wave/edge, 64 edges/block -> 50000 blocks ---
  decode_bce<<<50000, 256, 0, stream>>>(src, dst, nsrc, ndst, zbuf, acc);

  final_out<<<1, 32, 0, stream>>>(acc, out);
}